// Decoder_67662914781492
// MI455X (gfx1250) — compile-verified
//
#include <hip/hip_runtime.h>

// Attention-LSTM decoder for MI455X (gfx1250), fp32 end-to-end.
// GEMMs: V_WMMA_F32_16X16X4_F32, wave tile 32x64 (2 M-frags x 4 N-frags),
// register double-buffered so loads for step k+1 overlap WMMAs of step k.
// out_w (131MB) is L2-resident (192MB L2) -> logits GEMM is matrix-bound.

typedef __attribute__((ext_vector_type(2))) float v2f;
typedef __attribute__((ext_vector_type(8))) float v8f;

#define kB 128
#define kL 64
#define kT 20
#define kE 512
#define kH 1024
#define kA 512
#define kF 2048
#define kV 32000

#if defined(__HIP_DEVICE_COMPILE__) && \
    __has_builtin(__builtin_amdgcn_global_load_async_to_lds_b32)
#define HAVE_ASYNC_LDS 1
#else
#define HAVE_ASYNC_LDS 0
#endif

#if HAVE_ASYNC_LDS
typedef __attribute__((address_space(1))) int* gmem_i32_t;
typedef __attribute__((address_space(3))) int* lds_i32_t;
#endif

__device__ __forceinline__ v8f wmma_f32_k4(v2f a, v2f b, v8f c) {
  // 8 args: (neg_a, A, neg_b, B, c_mod, C, reuse_a, reuse_b)
  return __builtin_amdgcn_wmma_f32_16x16x4_f32(false, a, false, b, (short)0, c,
                                               false, false);
}

__device__ __forceinline__ void wmma8(v8f acc[2][4], const v2f a0, const v2f a1,
                                      const v2f b[4]) {
#pragma unroll
  for (int fi = 0; fi < 4; ++fi) {
    acc[0][fi] = wmma_f32_k4(a0, b[fi], acc[0][fi]);
    acc[1][fi] = wmma_f32_k4(a1, b[fi], acc[1][fi]);
  }
}

// ---------------------------------------------------------------------------
// fp32 WMMA GEMM: C[M,N] = A[M,K] @ B[K,N] + bias[N]
// grid = (N/256, M/32), block = 128 (4 waves), wave tile = 32x64.
// Exact tiling (M%32==0, N covered exactly, K%4==0) -> no masks, EXEC all-1s.
// ---------------------------------------------------------------------------
__global__ __launch_bounds__(128) void gemm_f32_wmma(
    const float* __restrict__ A, long long lda, const float* __restrict__ Bm,
    long long ldb, const float* __restrict__ bias, float* __restrict__ C,
    long long ldc, int K) {
  const int lane = threadIdx.x & 31;
  const int wave = threadIdx.x >> 5;
  const int lh = lane >> 4;  // lane half: K-pair select (A/B), +8 rows (C)
  const int lm = lane & 15;
  const int row0 = blockIdx.y * 32;
  const int col0 = blockIdx.x * 256 + wave * 64;

  const float* pA0 = A + (size_t)(row0 + lm) * lda + 2 * lh;
  const float* pA1 = pA0 + (size_t)16 * lda;
  const float* pB0 = Bm + (size_t)(2 * lh) * ldb + col0 + lm;
  const float* pB1 = pB0 + ldb;
  const long long bstep = 4 * ldb;

  v8f acc[2][4];
#pragma unroll
  for (int mf = 0; mf < 2; ++mf)
#pragma unroll
    for (int fi = 0; fi < 4; ++fi) acc[mf][fi] = v8f{};

  // prologue: load k-step 0
  v2f a0c, a1c, bc[4];
  a0c.x = pA0[0];
  a0c.y = pA0[1];
  a1c.x = pA1[0];
  a1c.y = pA1[1];
#pragma unroll
  for (int fi = 0; fi < 4; ++fi) {
    bc[fi].x = pB0[fi * 16];
    bc[fi].y = pB1[fi * 16];
  }
  pA0 += 4;
  pA1 += 4;
  pB0 += bstep;
  pB1 += bstep;

  // steady state: issue loads for k+1, then WMMAs for k, then rotate
  for (int k0 = 4; k0 < K; k0 += 4) {
    v2f a0n, a1n, bn[4];
    a0n.x = pA0[0];
    a0n.y = pA0[1];
    a1n.x = pA1[0];
    a1n.y = pA1[1];
#pragma unroll
    for (int fi = 0; fi < 4; ++fi) {
      bn[fi].x = pB0[fi * 16];
      bn[fi].y = pB1[fi * 16];
    }
    pA0 += 4;
    pA1 += 4;
    pB0 += bstep;
    pB1 += bstep;

    wmma8(acc, a0c, a1c, bc);

    a0c = a0n;
    a1c = a1n;
#pragma unroll
    for (int fi = 0; fi < 4; ++fi) bc[fi] = bn[fi];
  }
  wmma8(acc, a0c, a1c, bc);

#pragma unroll
  for (int mf = 0; mf < 2; ++mf)
#pragma unroll
    for (int fi = 0; fi < 4; ++fi) {
      const int n = col0 + fi * 16 + lm;
      const float bv = bias ? bias[n] : 0.0f;
#pragma unroll
      for (int v = 0; v < 8; ++v) {
        const int m = row0 + mf * 16 + v + 8 * lh;
        C[(size_t)m * ldc + n] = acc[mf][fi][v] + bv;
      }
    }
}

// ---------------------------------------------------------------------------
// LSTM gate GEMM: gates[128,4096] = [emb[y[:,t]], context] @ w_ih
//                                 + h @ w_hh + b_ih + b_hh
// Same 32x64 wave tiling, two pipelined phases. grid = (16, 4), block = 128.
// ---------------------------------------------------------------------------
__global__ __launch_bounds__(128) void lstm_gates_wmma(
    const float* __restrict__ emb, const int* __restrict__ y, int t,
    const float* __restrict__ context, const float* __restrict__ h,
    const float* __restrict__ w_ih, const float* __restrict__ w_hh,
    const float* __restrict__ b_ih, const float* __restrict__ b_hh,
    float* __restrict__ gates) {
  const int lane = threadIdx.x & 31;
  const int wave = threadIdx.x >> 5;
  const int lh = lane >> 4;
  const int lm = lane & 15;
  const int row0 = blockIdx.y * 32;
  const int col0 = blockIdx.x * 256 + wave * 64;
  const long long N4H = 4 * kH;

  const int b0 = row0 + lm, b1 = b0 + 16;
  const float* __restrict__ e0 = emb + (size_t)y[b0 * (kT + 1) + t] * kE;
  const float* __restrict__ e1 = emb + (size_t)y[b1 * (kT + 1) + t] * kE;
  const float* __restrict__ c0r = context + (size_t)b0 * kF;
  const float* __restrict__ c1r = context + (size_t)b1 * kF;

  v8f acc[2][4];
#pragma unroll
  for (int mf = 0; mf < 2; ++mf)
#pragma unroll
    for (int fi = 0; fi < 4; ++fi) acc[mf][fi] = v8f{};

  // A fetch for phase 1: concat [emb_row | context_row]; kE%4==0 so the
  // select is wave-uniform at every k-step (kk and kk+1 on the same side).
  auto ldA = [&](const float* __restrict__ er, const float* __restrict__ cr,
                 int kk) -> v2f {
    v2f a;
    if (kk < kE) {
      a.x = er[kk];
      a.y = er[kk + 1];
    } else {
      a.x = cr[kk - kE];
      a.y = cr[kk - kE + 1];
    }
    return a;
  };

  // ---- Phase 1: K = E + F over w_ih ----
  {
    const float* pB0 = w_ih + (size_t)(2 * lh) * N4H + col0 + lm;
    const float* pB1 = pB0 + N4H;
    v2f a0c = ldA(e0, c0r, 2 * lh), a1c = ldA(e1, c1r, 2 * lh), bc[4];
#pragma unroll
    for (int fi = 0; fi < 4; ++fi) {
      bc[fi].x = pB0[fi * 16];
      bc[fi].y = pB1[fi * 16];
    }
    pB0 += 4 * N4H;
    pB1 += 4 * N4H;
    for (int k0 = 4; k0 < kE + kF; k0 += 4) {
      v2f a0n = ldA(e0, c0r, k0 + 2 * lh), a1n = ldA(e1, c1r, k0 + 2 * lh);
      v2f bn[4];
#pragma unroll
      for (int fi = 0; fi < 4; ++fi) {
        bn[fi].x = pB0[fi * 16];
        bn[fi].y = pB1[fi * 16];
      }
      pB0 += 4 * N4H;
      pB1 += 4 * N4H;
      wmma8(acc, a0c, a1c, bc);
      a0c = a0n;
      a1c = a1n;
#pragma unroll
      for (int fi = 0; fi < 4; ++fi) bc[fi] = bn[fi];
    }
    wmma8(acc, a0c, a1c, bc);
  }

  // ---- Phase 2: K = H over w_hh ----
  {
    const float* pH0 = h + (size_t)b0 * kH + 2 * lh;
    const float* pH1 = h + (size_t)b1 * kH + 2 * lh;
    const float* pB0 = w_hh + (size_t)(2 * lh) * N4H + col0 + lm;
    const float* pB1 = pB0 + N4H;
    v2f a0c, a1c, bc[4];
    a0c.x = pH0[0];
    a0c.y = pH0[1];
    a1c.x = pH1[0];
    a1c.y = pH1[1];
#pragma unroll
    for (int fi = 0; fi < 4; ++fi) {
      bc[fi].x = pB0[fi * 16];
      bc[fi].y = pB1[fi * 16];
    }
    pH0 += 4;
    pH1 += 4;
    pB0 += 4 * N4H;
    pB1 += 4 * N4H;
    for (int k0 = 4; k0 < kH; k0 += 4) {
      v2f a0n, a1n, bn[4];
      a0n.x = pH0[0];
      a0n.y = pH0[1];
      a1n.x = pH1[0];
      a1n.y = pH1[1];
#pragma unroll
      for (int fi = 0; fi < 4; ++fi) {
        bn[fi].x = pB0[fi * 16];
        bn[fi].y = pB1[fi * 16];
      }
      pH0 += 4;
      pH1 += 4;
      pB0 += 4 * N4H;
      pB1 += 4 * N4H;
      wmma8(acc, a0c, a1c, bc);
      a0c = a0n;
      a1c = a1n;
#pragma unroll
      for (int fi = 0; fi < 4; ++fi) bc[fi] = bn[fi];
    }
    wmma8(acc, a0c, a1c, bc);
  }

#pragma unroll
  for (int mf = 0; mf < 2; ++mf)
#pragma unroll
    for (int fi = 0; fi < 4; ++fi) {
      const int n = col0 + fi * 16 + lm;
      const float bv = b_ih[n] + b_hh[n];
#pragma unroll
      for (int v = 0; v < 8; ++v) {
        const int m = row0 + mf * 16 + v + 8 * lh;
        gates[(size_t)m * (4 * kH) + n] = acc[mf][fi][v] + bv;
      }
    }
}

// ---------------------------------------------------------------------------
// Bahdanau attention. xd[b,:] and score_w are re-read 64x per block, so both
// are staged into LDS -- via gfx1250 async global->LDS loads when available.
// grid = B, block = 256.
// ---------------------------------------------------------------------------
__global__ __launch_bounds__(256) void attention_kernel(
    const float* __restrict__ X, const float* __restrict__ x_enc,
    const float* __restrict__ xd, const float* __restrict__ score_w,
    const float* __restrict__ score_b, float* __restrict__ context,
    float* __restrict__ weights, int t) {
  const int b = blockIdx.x;
  const int tid = threadIdx.x;
  const int lane = tid & 31;
  const int wave = tid >> 5;
  __shared__ float s_xd[kA];
  __shared__ float s_sw[kA];
  __shared__ float sc[kL];

#if HAVE_ASYNC_LDS
  for (int i = tid; i < kA; i += 256) {
    __builtin_amdgcn_global_load_async_to_lds_b32(
        (gmem_i32_t)(xd + (size_t)b * kA + i), (lds_i32_t)(&s_xd[i]), 0, 0);
    __builtin_amdgcn_global_load_async_to_lds_b32(
        (gmem_i32_t)(score_w + i), (lds_i32_t)(&s_sw[i]), 0, 0);
  }
  asm volatile("s_wait_asynccnt 0" ::: "memory");
#else
  for (int i = tid; i < kA; i += 256) {
    s_xd[i] = xd[(size_t)b * kA + i];
    s_sw[i] = score_w[i];
  }
#endif
  __syncthreads();

  // scores: each wave handles 8 of the L=64 positions, lanes stride A=512
  for (int l = wave * 8; l < wave * 8 + 8; ++l) {
    const float* __restrict__ xe = x_enc + ((size_t)b * kL + l) * kA;
    float p = 0.0f;
    for (int a = lane; a < kA; a += 32) p += tanhf(xe[a] + s_xd[a]) * s_sw[a];
#pragma unroll
    for (int off = 16; off; off >>= 1) p += __shfl_xor(p, off, 32);
    if (lane == 0) sc[l] = p + score_b[0];
  }
  __syncthreads();

  // softmax over 64 by wave 0 (2 values per lane)
  if (wave == 0) {
    float s0 = sc[lane], s1 = sc[lane + 32];
    float mx = fmaxf(s0, s1);
#pragma unroll
    for (int off = 16; off; off >>= 1) mx = fmaxf(mx, __shfl_xor(mx, off, 32));
    float e0 = __expf(s0 - mx), e1 = __expf(s1 - mx);
    float sum = e0 + e1;
#pragma unroll
    for (int off = 16; off; off >>= 1) sum += __shfl_xor(sum, off, 32);
    const float inv = 1.0f / sum;
    sc[lane] = e0 * inv;
    sc[lane + 32] = e1 * inv;
    float* wrow = weights + ((size_t)b * kT + t) * kL;
    wrow[lane] = e0 * inv;
    wrow[lane + 32] = e1 * inv;
  }
  __syncthreads();

  // context[b,f] = sum_l alpha[l] * X[b,l,f]
  for (int f = tid; f < kF; f += 256) {
    float s = 0.0f;
#pragma unroll 8
    for (int l = 0; l < kL; ++l) s += sc[l] * X[((size_t)b * kL + l) * kF + f];
    context[(size_t)b * kF + f] = s;
  }
}

__global__ __launch_bounds__(256) void lstm_pointwise(
    const float* __restrict__ gates, const float* __restrict__ c_in,
    float* __restrict__ h_out, float* __restrict__ c_out) {
  const int i = blockIdx.x * 256 + threadIdx.x;  // over B*H
  const int b = i / kH, j = i % kH;
  const float* __restrict__ g = gates + (size_t)b * 4 * kH;
  float ig = g[j], fg = g[kH + j], gg = g[2 * kH + j], og = g[3 * kH + j];
  ig = 1.0f / (1.0f + __expf(-ig));
  fg = 1.0f / (1.0f + __expf(-fg));
  og = 1.0f / (1.0f + __expf(-og));
  const float c2 = fg * c_in[i] + ig * tanhf(gg);
  c_out[i] = c2;
  h_out[i] = og * tanhf(c2);
}

__global__ __launch_bounds__(256) void mean_pool(const float* __restrict__ X,
                                                 float* __restrict__ mean_ann) {
  const int i = blockIdx.x * 256 + threadIdx.x;  // over B*F
  const int b = i / kF, f = i % kF;
  float s = 0.0f;
#pragma unroll 8
  for (int l = 0; l < kL; ++l) s += X[((size_t)b * kL + l) * kF + f];
  mean_ann[i] = s * (1.0f / kL);
}

extern "C" void kernel_launch(void* const* d_in, const int* in_sizes, int n_in,
                              void* d_out, int out_size, void* d_ws,
                              size_t ws_size, hipStream_t stream) {
  (void)in_sizes; (void)n_in; (void)out_size; (void)ws_size;
  const float* X       = (const float*)d_in[0];
  const int*   y       = (const int*)d_in[1];
  const float* emb     = (const float*)d_in[2];
  const float* fc1_w   = (const float*)d_in[3];
  const float* fc1_b   = (const float*)d_in[4];
  const float* fc2_w   = (const float*)d_in[5];
  const float* fc2_b   = (const float*)d_in[6];
  const float* score_w = (const float*)d_in[7];
  const float* score_b = (const float*)d_in[8];
  const float* sm_w    = (const float*)d_in[9];
  const float* sm_b    = (const float*)d_in[10];
  const float* lm_w    = (const float*)d_in[11];
  const float* lm_b    = (const float*)d_in[12];
  const float* w_ih    = (const float*)d_in[13];
  const float* b_ih    = (const float*)d_in[14];
  const float* w_hh    = (const float*)d_in[15];
  const float* b_hh    = (const float*)d_in[16];
  const float* out_w   = (const float*)d_in[17];
  const float* out_b   = (const float*)d_in[18];

  // Workspace layout (floats); total ~5.84M floats (~23.4 MB)
  float* ws = (float*)d_ws;
  float* mean_ann = ws;            ws += (size_t)kB * kF;
  float* x_enc    = ws;            ws += (size_t)kB * kL * kA;
  float* xd       = ws;            ws += (size_t)kB * kA;
  float* ctx      = ws;            ws += (size_t)kB * kF;
  float* gates    = ws;            ws += (size_t)kB * 4 * kH;
  float* hA       = ws;            ws += (size_t)kB * kH;
  float* cA       = ws;            ws += (size_t)kB * kH;
  float* hB       = ws;            ws += (size_t)kB * kH;
  float* cB       = ws;            ws += (size_t)kB * kH;

  float* outputs = (float*)d_out;                   // [B,T,V]
  float* weights = outputs + (size_t)kB * kT * kV;  // [B,T,L]

  const dim3 blk(128);

  // mean_ann = X.mean(axis=1)
  mean_pool<<<(kB * kF) / 256, 256, 0, stream>>>(X, mean_ann);
  // h0 = mean_ann @ sm_w + sm_b ; c0 = mean_ann @ lm_w + lm_b (M=128,K=2048,N=1024)
  gemm_f32_wmma<<<dim3(kH / 256, kB / 32), blk, 0, stream>>>(
      mean_ann, kF, sm_w, kH, sm_b, hA, kH, kF);
  gemm_f32_wmma<<<dim3(kH / 256, kB / 32), blk, 0, stream>>>(
      mean_ann, kF, lm_w, kH, lm_b, cA, kH, kF);
  // x_enc = X @ fc2_w + fc2_b  (M=8192,K=2048,N=512)
  gemm_f32_wmma<<<dim3(kA / 256, (kB * kL) / 32), blk, 0, stream>>>(
      X, kF, fc2_w, kA, fc2_b, x_enc, kA, kF);

  float *h = hA, *c = cA, *h2 = hB, *c2 = cB;
  for (int t = 0; t < kT; ++t) {
    // xd = h @ fc1_w + fc1_b  (M=128,K=1024,N=512)
    gemm_f32_wmma<<<dim3(kA / 256, kB / 32), blk, 0, stream>>>(
        h, kH, fc1_w, kA, fc1_b, xd, kA, kH);
    // attention -> ctx, alpha -> weights
    attention_kernel<<<kB, 256, 0, stream>>>(X, x_enc, xd, score_w, score_b,
                                             ctx, weights, t);
    // gates = [emb_t, ctx] @ w_ih + h @ w_hh + biases  (N=4096)
    lstm_gates_wmma<<<dim3((4 * kH) / 256, kB / 32), blk, 0, stream>>>(
        emb, y, t, ctx, h, w_ih, w_hh, b_ih, b_hh, gates);
    // (h2, c2) from gates, c
    lstm_pointwise<<<(kB * kH) / 256, 256, 0, stream>>>(gates, c, h2, c2);
    // logits = h2 @ out_w + out_b, written at outputs[:, t, :] via ldc = T*V
    gemm_f32_wmma<<<dim3(kV / 256, kB / 32), blk, 0, stream>>>(
        h2, kH, out_w, kV, out_b, outputs + (size_t)t * kV,
        (long long)kT * kV, kH);
    // carry swap
    float* tp;
    tp = h; h = h2; h2 = tp;
    tp = c; c = c2; c2 = tp;
  }
}